// CondensateToPseudoRS_13408887898337
// MI455X (gfx1250) — compile-verified
//
#include <hip/hip_runtime.h>
#include <stdint.h>

#define TPB 1024
#define RADIUS2 2.25f   // 1.5^2
#define THRESH  0.8f

// ---------------------------------------------------------------------------
// gfx1250 feature guards (fall back gracefully if a builtin is absent)
// ---------------------------------------------------------------------------
#if defined(__HIP_DEVICE_COMPILE__)
#if __has_builtin(__builtin_amdgcn_tensor_load_to_lds) && \
    __has_builtin(__builtin_amdgcn_s_wait_tensorcnt)
#define HAVE_TDM 1
#endif
#if __has_builtin(__builtin_amdgcn_global_load_async_to_lds_b128) && \
    __has_builtin(__builtin_amdgcn_s_wait_asynccnt)
#define HAVE_ASYNC_LDS 1
#endif
#if __has_builtin(__builtin_amdgcn_global_store_async_from_lds_b128) && \
    __has_builtin(__builtin_amdgcn_s_wait_asynccnt)
#define HAVE_ASYNC_STORE 1
#endif
#endif

#if defined(HAVE_TDM) || defined(HAVE_ASYNC_LDS) || defined(HAVE_ASYNC_STORE)
// Matches the async builtins' parameter type: 'int __attribute__((vector_size(16)))'
typedef int async_v4i __attribute__((vector_size(16)));
typedef __attribute__((address_space(1))) async_v4i* async_gptr;
typedef __attribute__((address_space(3))) async_v4i* async_lptr;
#endif

#if defined(HAVE_TDM)
// TDM descriptor groups (clang-23 6-arg form: u32x4, i32x8, i32x4, i32x4, i32x8, i32)
typedef unsigned tdm_u32x4 __attribute__((vector_size(16)));
typedef int      tdm_i32x8 __attribute__((vector_size(32)));
typedef int      tdm_i32x4 __attribute__((vector_size(16)));
#endif

// ---------------------------------------------------------------------------
// 0) init: asso = -1, psrs padding = N
// ---------------------------------------------------------------------------
__global__ __launch_bounds__(TPB) void k_init(int* __restrict__ asso,
                                              float* __restrict__ psrs, int n) {
  int i = blockIdx.x * TPB + threadIdx.x;
  if (i < n) asso[i] = -1;
  if (i < n + 1) psrs[i] = (float)n;
}

// ---------------------------------------------------------------------------
// 1) Greedy condensation. Single persistent workgroup (32 wave32 waves).
//    Fuses "assign points to previous pick" with "argmax for next pick" in a
//    single sweep per iteration. Data (asso/betas/ccoords ~4MB) is L2-resident.
// ---------------------------------------------------------------------------
__global__ __launch_bounds__(TPB) void k_greedy(const float* __restrict__ cc,
                                                const float* __restrict__ beta,
                                                const int* __restrict__ rs, int nrs,
                                                int* __restrict__ asso, int n, int D) {
  __shared__ float s_m[TPB];
  __shared__ int   s_i[TPB];
  __shared__ float s_ck[8];
  __shared__ int   s_k;
  __shared__ int   s_segk;
  __shared__ int   s_rs[64];

  const int tid = threadIdx.x;
  const int ninter = (nrs > 2) ? (nrs - 2) : 0;
  if (tid < nrs && tid < 64) s_rs[tid] = rs[tid];
  __syncthreads();

  int   k = -1, segk = 0;
  float ck[8];
  for (int t = 0; t < 8; ++t) ck[t] = 0.f;

  for (int iter = 0; iter <= n; ++iter) {
    float bm = -1e30f;
    int   bi = -1;
    for (int i = tid; i < n; i += TPB) {
      if (i + TPB < n) __builtin_prefetch(&asso[i + TPB], 0, 1);  // global_prefetch_b8
      int a = asso[i];
      if (a >= 0) continue;
      int s = 0;
      for (int t = 0; t < ninter; ++t) s += (i >= s_rs[t + 1]) ? 1 : 0;
      if (k >= 0 && s == segk) {
        float d2 = 0.f;
        for (int t = 0; t < D; ++t) {
          float dd = cc[(size_t)i * D + t] - ck[t];
          d2 += dd * dd;
        }
        if (d2 <= RADIUS2) { asso[i] = k; continue; }  // assign to previous pick
      }
      float b = beta[i];
      if (b > bm) { bm = b; bi = i; }  // strided ascending -> first-index ties
    }
    s_m[tid] = bm; s_i[tid] = bi;
    __syncthreads();
    for (int off = TPB >> 1; off > 0; off >>= 1) {
      if (tid < off) {
        float mo = s_m[tid + off]; int io = s_i[tid + off];
        float mc = s_m[tid];       int ic = s_i[tid];
        bool take = (mo > mc) || (mo == mc && io >= 0 && (ic < 0 || io < ic));
        if (take) { s_m[tid] = mo; s_i[tid] = io; }
      }
      __syncthreads();
    }
    if (tid == 0) {
      int kn = (s_i[0] >= 0 && s_m[0] >= THRESH) ? s_i[0] : -2;
      s_k = kn;
      if (kn >= 0) {
        for (int t = 0; t < D; ++t) s_ck[t] = cc[(size_t)kn * D + t];
        int s = 0;
        for (int t = 0; t < ninter; ++t) s += (kn >= s_rs[t + 1]) ? 1 : 0;
        s_segk = s;
      }
    }
    __syncthreads();
    k = s_k;
    if (k < 0) break;
    for (int t = 0; t < D; ++t) ck[t] = s_ck[t];
    segk = s_segk;
    __syncthreads();
  }
}

// ---------------------------------------------------------------------------
// 2) keys/vals for sort + asso output (original order, as float)
// ---------------------------------------------------------------------------
__global__ __launch_bounds__(TPB) void k_pairs(const int* __restrict__ asso,
                                               unsigned* __restrict__ keys,
                                               int* __restrict__ vals,
                                               float* __restrict__ asso_f, int n) {
  int i = blockIdx.x * TPB + threadIdx.x;
  if (i < n) {
    int a = asso[i];
    keys[i]  = (unsigned)(a + 1);  // -1 -> 0, cpoint k -> k+1 (fits in 18 bits)
    vals[i]  = i;
    asso_f[i] = (float)a;
  }
}

// ---------------------------------------------------------------------------
// 3) Stable LSD radix sort: per-block digit histograms (digit-major layout)
// ---------------------------------------------------------------------------
__global__ __launch_bounds__(TPB) void k_hist(const unsigned* __restrict__ keys, int n,
                                              int shift, unsigned* __restrict__ hist,
                                              int nblocks) {
  __shared__ unsigned h[256];
  const int tid = threadIdx.x;
  if (tid < 256) h[tid] = 0u;
  __syncthreads();
  const int i = blockIdx.x * TPB + tid;
  if (i < n) {
    unsigned d = (keys[i] >> shift) & 255u;
    atomicAdd(&h[d], 1u);
  }
  __syncthreads();
  if (tid < 256) hist[(size_t)tid * nblocks + blockIdx.x] = h[tid];
}

// Single-block exclusive scan over an arbitrary-length unsigned array.
__global__ __launch_bounds__(TPB) void k_exscan(unsigned* __restrict__ a, int L) {
  __shared__ unsigned s[TPB];
  __shared__ unsigned carry;
  if (threadIdx.x == 0) carry = 0u;
  __syncthreads();
  for (int base = 0; base < L; base += TPB) {
    const int i = base + threadIdx.x;
    unsigned v = (i < L) ? a[i] : 0u;
    s[threadIdx.x] = v;
    __syncthreads();
    for (int off = 1; off < TPB; off <<= 1) {
      unsigned t = (threadIdx.x >= (unsigned)off) ? s[threadIdx.x - off] : 0u;
      __syncthreads();
      s[threadIdx.x] += t;
      __syncthreads();
    }
    unsigned c = carry;
    if (i < L) a[i] = c + s[threadIdx.x] - v;  // exclusive
    __syncthreads();
    if (threadIdx.x == 0) carry = c + s[TPB - 1];
    __syncthreads();
  }
}

// Stable scatter: wave32 ballot-based same-digit ranking + cross-wave LDS scan.
__global__ __launch_bounds__(TPB) void k_scatter(const unsigned* __restrict__ keys,
                                                 const int* __restrict__ vals,
                                                 int n, int shift,
                                                 const unsigned* __restrict__ offs,
                                                 int nblocks,
                                                 unsigned* __restrict__ okeys,
                                                 int* __restrict__ ovals) {
  __shared__ unsigned whist[32 * 256];
  const int tid  = threadIdx.x;
  const int lane = tid & 31;
  const int wave = tid >> 5;
  const int b    = blockIdx.x;
  const int i    = b * TPB + tid;
  const bool valid = (i < n);

  unsigned key   = valid ? keys[i] : 0xFFFFFFFFu;
  unsigned digit = (key >> shift) & 255u;

  unsigned vmask = (unsigned)__ballot(valid ? 1 : 0);
  unsigned m = 0xFFFFFFFFu;
  for (int bit = 0; bit < 8; ++bit) {
    unsigned vb = (unsigned)__ballot((int)((digit >> bit) & 1u));
    m &= ((digit >> bit) & 1u) ? vb : ~vb;
  }
  m &= vmask;
  unsigned below = m & ((1u << lane) - 1u);
  unsigned rank  = (unsigned)__popc(below);
  unsigned cnt   = (unsigned)__popc(m);

  for (int j = tid; j < 32 * 256; j += TPB) whist[j] = 0u;
  __syncthreads();
  if (valid && rank == 0u) whist[wave * 256 + (int)digit] = cnt;  // digit leader
  __syncthreads();
  if (tid < 256) {  // exclusive prefix across waves, per digit
    unsigned run = 0u;
    for (int w = 0; w < 32; ++w) {
      unsigned c = whist[w * 256 + tid];
      whist[w * 256 + tid] = run;
      run += c;
    }
  }
  __syncthreads();
  if (valid) {
    unsigned pos = offs[(size_t)digit * nblocks + b] + whist[wave * 256 + (int)digit] + rank;
    okeys[pos] = key;
    ovals[pos] = vals[i];
  }
}

// ---------------------------------------------------------------------------
// 4) Segment boundaries -> belongs (inclusive scan) + psrs starts + sids
// ---------------------------------------------------------------------------
__global__ __launch_bounds__(TPB) void k_flags(const unsigned* __restrict__ skeys, int n,
                                               unsigned* __restrict__ parts) {
  __shared__ unsigned red[TPB];
  const int j = blockIdx.x * TPB + threadIdx.x;
  unsigned f = 0u;
  if (j > 0 && j < n) f = (skeys[j] != skeys[j - 1]) ? 1u : 0u;
  red[threadIdx.x] = f;
  __syncthreads();
  for (int off = TPB >> 1; off > 0; off >>= 1) {
    if ((int)threadIdx.x < off) red[threadIdx.x] += red[threadIdx.x + off];
    __syncthreads();
  }
  if (threadIdx.x == 0) parts[blockIdx.x] = red[0];
}

__global__ __launch_bounds__(TPB) void k_finalize(const unsigned* __restrict__ skeys,
                                                  const int* __restrict__ svals,
                                                  const unsigned* __restrict__ parts,
                                                  int n,
                                                  float* __restrict__ psrs,
                                                  float* __restrict__ sids,
                                                  float* __restrict__ belongs) {
  __shared__ unsigned s[TPB];
  const int j = blockIdx.x * TPB + threadIdx.x;
  unsigned f = 0u;
  if (j > 0 && j < n) f = (skeys[j] != skeys[j - 1]) ? 1u : 0u;
  s[threadIdx.x] = f;
  __syncthreads();
  for (int off = 1; off < TPB; off <<= 1) {  // inclusive scan of flags
    unsigned t = (threadIdx.x >= (unsigned)off) ? s[threadIdx.x - off] : 0u;
    __syncthreads();
    s[threadIdx.x] += t;
    __syncthreads();
  }
  if (j < n) {
    unsigned bseg = parts[blockIdx.x] + s[threadIdx.x];
    belongs[j] = (float)bseg;
    sids[j]    = (float)svals[j];
    if (j == 0)      psrs[0]    = 0.0f;
    else if (f)      psrs[bseg] = (float)j;  // first element of each segment
  }
}

// ---------------------------------------------------------------------------
// 5) Row gather: sdata[j] = data[order[j]].
//    8 waves/block, 8 rows per wave (64 rows per block on every path).
//    Preferred path: TDM gather-mode tensor_load_to_lds (8 x 32-bit row
//    indices per descriptor) -> s_wait_tensorcnt -> async store from LDS.
// ---------------------------------------------------------------------------
__global__ __launch_bounds__(256) void k_gather(const float* __restrict__ data,
                                                const int* __restrict__ order,
                                                float* __restrict__ out, int n, int F) {
  const int lane = threadIdx.x & 31;
  const int wave = threadIdx.x >> 5;
  const int base = blockIdx.x * 64 + wave * 8;   // first row for this wave
  if (base >= n) return;
  int nrows = n - base;
  if (nrows > 8) nrows = 8;

#if defined(HAVE_TDM)
  __shared__ float4 stage[8][256];   // 8 waves x (8 rows x 128 floats)

  int idx[8];
  for (int j = 0; j < 8; ++j) idx[j] = (j < nrows) ? order[base + j] : 0;

  unsigned lds_off =
      (unsigned)(size_t)(__attribute__((address_space(3))) float4*)&stage[wave][0];
  unsigned long long ga = (unsigned long long)(size_t)data;

  // D# group 0: count=1, gather_mode=1, 32-bit indices, lds_addr, global_addr, type=2
  tdm_u32x4 g0;
  g0[0] = 1u | (1u << 30) | (1u << 31);
  g0[1] = lds_off;
  g0[2] = (unsigned)(ga & 0xFFFFFFFFu);
  g0[3] = (unsigned)((ga >> 32) & 0x01FFFFFFu) | (2u << 30);

  // D# group 1: data_size=4B; tensor_dim0=F; tensor_dim1=n; tile_dim0=F;
  //             tile_dim1=#valid indices; tensor_dim0_stride=F
  tdm_i32x8 g1;
  g1[0] = (2 << 16);                                      // data_size code 2 (4B)
  g1[1] = (F & 0xFFFF) << 16;                             // tensor_dim0[15:0]
  g1[2] = ((F >> 16) & 0xFFFF) | ((n & 0xFFFF) << 16);    // dim0 hi | dim1 lo
  g1[3] = ((n >> 16) & 0xFFFF) | ((F & 0xFFFF) << 16);    // dim1 hi | tile_dim0
  g1[4] = nrows;                                          // tile_dim1 (valid idx count)
  g1[5] = F;                                              // tensor_dim0_stride[31:0]
  g1[6] = 0;
  g1[7] = 0;

  // D# groups 2/3: eight 32-bit row indices
  tdm_i32x4 g2; g2[0] = idx[0]; g2[1] = idx[1]; g2[2] = idx[2]; g2[3] = idx[3];
  tdm_i32x4 g3; g3[0] = idx[4]; g3[1] = idx[5]; g3[2] = idx[6]; g3[3] = idx[7];

  // clang-23 toolchain takes a 6-arg form with an extra (reserved) i32x8 group.
  tdm_i32x8 gx = {0, 0, 0, 0, 0, 0, 0, 0};

  __builtin_amdgcn_tensor_load_to_lds(g0, g1, g2, g3, gx, 0);
  __builtin_amdgcn_s_wait_tensorcnt(0);

  // Rows are compacted in LDS; output rows base..base+nrows-1 are contiguous.
  float4* dst = (float4*)(out + (size_t)base * F);
  const int nv = nrows * (F >> 2);
#if defined(HAVE_ASYNC_STORE)
  for (int t = lane; t < nv; t += 32) {
    __builtin_amdgcn_global_store_async_from_lds_b128(
        (async_gptr)(dst + t), (async_lptr)&stage[wave][t], 0, 0);
  }
  __builtin_amdgcn_s_wait_asynccnt(0);
#else
  for (int t = lane; t < nv; t += 32) dst[t] = stage[wave][t];
#endif

#elif defined(HAVE_ASYNC_LDS)
  __shared__ float4 stage[8][32];
  for (int r = 0; r < nrows; ++r) {
    const int row = base + r;
    const int src = order[row];
    const float4* s = (const float4*)(data + (size_t)src * F);
    float4* d       = (float4*)(out + (size_t)row * F);
    for (int c = lane; c < (F >> 2); c += 32) {
      __builtin_amdgcn_global_load_async_to_lds_b128(
          (async_gptr)(s + c), (async_lptr)&stage[wave][lane], 0, 0);
      __builtin_amdgcn_s_wait_asynccnt(0);
      d[c] = stage[wave][lane];
    }
  }
#else
  for (int r = 0; r < nrows; ++r) {
    const int row = base + r;
    const int src = order[row];
    const float4* s = (const float4*)(data + (size_t)src * F);
    float4* d       = (float4*)(out + (size_t)row * F);
    for (int c = lane; c < (F >> 2); c += 32) d[c] = s[c];
  }
#endif
}

// ---------------------------------------------------------------------------
extern "C" void kernel_launch(void* const* d_in, const int* in_sizes, int n_in,
                              void* d_out, int out_size, void* d_ws, size_t ws_size,
                              hipStream_t stream) {
  const float* data  = (const float*)d_in[0];
  const float* cc    = (const float*)d_in[1];
  const float* betas = (const float*)d_in[2];
  const int*   rs    = (const int*)d_in[3];

  const int n   = in_sizes[2];            // betas is [N,1]
  const int F   = in_sizes[0] / n;        // 128
  const int D   = in_sizes[1] / n;        // 3
  const int nrs = in_sizes[3];            // 3

  float* out       = (float*)d_out;
  float* o_sdata   = out;                         // N*F
  float* o_psrs    = out + (size_t)n * F;         // N+1
  float* o_sids    = o_psrs + (n + 1);            // N
  float* o_asso    = o_sids + n;                  // N
  float* o_belongs = o_asso + n;                  // N

  const int B = (n + TPB - 1) / TPB;

  char* w = (char*)d_ws;
  auto take = [&](size_t bytes) {
    char* p = w;
    w += (bytes + 255) & ~(size_t)255;
    return p;
  };
  int*      asso  = (int*)take((size_t)n * 4);
  unsigned* keysA = (unsigned*)take((size_t)n * 4);
  int*      valsA = (int*)take((size_t)n * 4);
  unsigned* keysB = (unsigned*)take((size_t)n * 4);
  int*      valsB = (int*)take((size_t)n * 4);
  unsigned* hist  = (unsigned*)take((size_t)256 * B * 4);
  unsigned* parts = (unsigned*)take((size_t)B * 4);

  k_init<<<(n + 1 + TPB - 1) / TPB, TPB, 0, stream>>>(asso, o_psrs, n);
  k_greedy<<<1, TPB, 0, stream>>>(cc, betas, rs, nrs, asso, n, D);
  k_pairs<<<B, TPB, 0, stream>>>(asso, keysA, valsA, o_asso, n);

  unsigned* ck = keysA; int* cv = valsA;
  unsigned* nk = keysB; int* nv = valsB;
  for (int shift = 0; shift < 24; shift += 8) {   // 3 x 8-bit stable passes
    k_hist<<<B, TPB, 0, stream>>>(ck, n, shift, hist, B);
    k_exscan<<<1, TPB, 0, stream>>>(hist, 256 * B);
    k_scatter<<<B, TPB, 0, stream>>>(ck, cv, n, shift, hist, B, nk, nv);
    unsigned* tk = ck; ck = nk; nk = tk;
    int* tv = cv; cv = nv; nv = tv;
  }
  // sorted keys in ck, sorted original indices (order) in cv

  k_flags<<<B, TPB, 0, stream>>>(ck, n, parts);
  k_exscan<<<1, TPB, 0, stream>>>(parts, B);
  k_finalize<<<B, TPB, 0, stream>>>(ck, cv, parts, n, o_psrs, o_sids, o_belongs);
  k_gather<<<(n + 63) / 64, 256, 0, stream>>>(data, cv, o_sdata, n, F);
}